// ElectrostaticPotential_82910048682170
// MI455X (gfx1250) — compile-verified
//
#include <hip/hip_runtime.h>
#include <math.h>

#define B_   4
#define NC_  128
#define NM_  16384
#define NPAIR 127
#define OSTRIDE 112   // padded row stride for `out` (100 cols used)
#define OUTC 1700     // 100 esp + 1600 G per (b,n)

typedef __attribute__((ext_vector_type(2))) float v2f;
typedef __attribute__((ext_vector_type(8))) float v8f;

// ---------------------------------------------------------------------------
// K1: per (b,n) reduce over NM charges -> esp scalar + efield 3-vector
// ---------------------------------------------------------------------------
__global__ __launch_bounds__(256) void k_esp_field(
    const float* __restrict__ ac,      // (B,NC,3)
    const float* __restrict__ cc,      // (B,NM,3)
    const float* __restrict__ q,       // (B,NM)
    const unsigned char* __restrict__ mask, // (B,NM) bool
    float* __restrict__ esp,           // (B*NC)
    float* __restrict__ ef)            // (B*NC,3)
{
    const float FACTOR = (float)(27.2114 * 0.529177249);
    int bn = blockIdx.x;
    int b  = bn >> 7;
    int tid = threadIdx.x;

    float axx = ac[bn * 3 + 0];
    float ayy = ac[bn * 3 + 1];
    float azz = ac[bn * 3 + 2];

    const float* ccb = cc + (size_t)b * NM_ * 3;
    const float* qb  = q  + (size_t)b * NM_;
    const unsigned char* mb = mask + (size_t)b * NM_;

    float se = 0.f, sx = 0.f, sy = 0.f, sz = 0.f;
    for (int m = tid; m < NM_; m += 256) {
        float dx = axx - ccb[m * 3 + 0];
        float dy = ayy - ccb[m * 3 + 1];
        float dz = azz - ccb[m * 3 + 2];
        float d2 = dx * dx + dy * dy + dz * dz;
        float d  = sqrtf(d2);
        float ds = fmaxf(d, 1e-12f);
        float inv  = mb[m] ? (1.0f / ds) : 0.0f;
        float inv3 = inv * inv * inv;
        float qq = qb[m];
        se += qq * inv;
        sx += qq * dx * inv3;
        sy += qq * dy * inv3;
        sz += qq * dz * inv3;
    }

    __shared__ float red[4][256];
    red[0][tid] = se; red[1][tid] = sx; red[2][tid] = sy; red[3][tid] = sz;
    __syncthreads();
    for (int s = 128; s > 0; s >>= 1) {
        if (tid < s) {
            red[0][tid] += red[0][tid + s];
            red[1][tid] += red[1][tid + s];
            red[2][tid] += red[2][tid + s];
            red[3][tid] += red[3][tid + s];
        }
        __syncthreads();
    }
    if (tid == 0) {
        esp[bn]        = red[0][0] * FACTOR;
        ef[bn * 3 + 0] = red[1][0] * FACTOR;
        ef[bn * 3 + 1] = red[2][0] * FACTOR;
        ef[bn * 3 + 2] = red[3][0] * FACTOR;
    }
}

// ---------------------------------------------------------------------------
// K2: fused proj + per-pair channel-indexed MLP (1->25->50->100, tanh,
// concat-residual on layers 1,2). One thread per neighbor slot; h0/h1 live in
// LDS as per-thread scratch (no cross-thread sharing -> no barriers needed).
// Row 127 (K-pad for the WMMA stage) is written as zeros.
// ---------------------------------------------------------------------------
__global__ __launch_bounds__(128) void k_ef_mlp(
    const float* __restrict__ ac,      // (B,NC,3)
    const float* __restrict__ ef,      // (B*NC,3)
    const float* __restrict__ W0, const float* __restrict__ b0,   // (36,25,1),(36,25)
    const float* __restrict__ W1, const float* __restrict__ b1,   // (36,50,25),(36,50)
    const float* __restrict__ W2, const float* __restrict__ b2,   // (36,100,50),(36,100)
    const int* __restrict__ types,     // (NC)
    float* __restrict__ outbuf)        // (B*NC, 128, OSTRIDE)
{
    __shared__ float sh0[128 * 25];
    __shared__ float sh1[128 * 50];

    int bn = blockIdx.x;
    int b  = bn >> 7;
    int n  = bn & 127;
    int i  = threadIdx.x;

    float* orow = outbuf + ((size_t)bn * 128 + i) * OSTRIDE;

    if (i < NPAIR) {
        int j  = (i < n) ? i : i + 1;
        int ch = types[n] * 6 + types[j];

        float rx = ac[bn * 3 + 0] - ac[(b * NC_ + j) * 3 + 0];
        float ry = ac[bn * 3 + 1] - ac[(b * NC_ + j) * 3 + 1];
        float rz = ac[bn * 3 + 2] - ac[(b * NC_ + j) * 3 + 2];
        float d2 = rx * rx + ry * ry + rz * rz;
        float pr = (rx * ef[bn * 3 + 0] + ry * ef[bn * 3 + 1] + rz * ef[bn * 3 + 2]) / d2;

        float* h0 = &sh0[i * 25];
        float* h1 = &sh1[i * 50];

        const float* w0  = W0 + ch * 25;
        const float* bb0 = b0 + ch * 25;
        for (int o = 0; o < 25; ++o)
            h0[o] = tanhf(fmaf(pr, w0[o], bb0[o]));

        const float* w1  = W1 + (size_t)ch * 50 * 25;
        const float* bb1 = b1 + ch * 50;
        for (int o = 0; o < 50; ++o) {
            float t = bb1[o];
            const float* wr = w1 + o * 25;
            #pragma unroll 5
            for (int k = 0; k < 25; ++k) t = fmaf(h0[k], wr[k], t);
            h1[o] = tanhf(t) + h0[(o < 25) ? o : o - 25];
        }

        const float* w2  = W2 + (size_t)ch * 100 * 50;
        const float* bb2 = b2 + ch * 100;
        for (int o = 0; o < 100; ++o) {
            float t = bb2[o];
            const float* wr = w2 + o * 50;
            #pragma unroll 5
            for (int k = 0; k < 50; ++k) t = fmaf(h1[k], wr[k], t);
            orow[o] = tanhf(t) + h1[(o < 50) ? o : o - 50];
        }
    } else {
        // zero K-padding row so the WMMA K=128 loop is exact
        for (int o = 0; o < 100; ++o) orow[o] = 0.0f;
    }
}

// ---------------------------------------------------------------------------
// K3: esp MLP (512 rows, channel = atom type), writes output cols [0,100)
// ---------------------------------------------------------------------------
__global__ __launch_bounds__(128) void k_esp_mlp(
    const float* __restrict__ esp,     // (B*NC)
    const float* __restrict__ W0, const float* __restrict__ b0,
    const float* __restrict__ W1, const float* __restrict__ b1,
    const float* __restrict__ W2, const float* __restrict__ b2,
    const int* __restrict__ types,
    float* __restrict__ dout)          // (B*NC, OUTC)
{
    __shared__ float sh0[128 * 25];
    __shared__ float sh1[128 * 50];

    int t  = blockIdx.x * 128 + threadIdx.x;   // bn in [0,512)
    int n  = t & 127;
    int ch = types[n];
    float x = esp[t];

    float* h0 = &sh0[threadIdx.x * 25];
    float* h1 = &sh1[threadIdx.x * 50];
    float* orow = dout + (size_t)t * OUTC;

    const float* w0  = W0 + ch * 25;
    const float* bb0 = b0 + ch * 25;
    for (int o = 0; o < 25; ++o)
        h0[o] = tanhf(fmaf(x, w0[o], bb0[o]));

    const float* w1  = W1 + (size_t)ch * 50 * 25;
    const float* bb1 = b1 + ch * 50;
    for (int o = 0; o < 50; ++o) {
        float s = bb1[o];
        const float* wr = w1 + o * 25;
        #pragma unroll 5
        for (int k = 0; k < 25; ++k) s = fmaf(h0[k], wr[k], s);
        h1[o] = tanhf(s) + h0[(o < 25) ? o : o - 25];
    }

    const float* w2  = W2 + (size_t)ch * 100 * 50;
    const float* bb2 = b2 + ch * 100;
    for (int o = 0; o < 100; ++o) {
        float s = bb2[o];
        const float* wr = w2 + o * 50;
        #pragma unroll 5
        for (int k = 0; k < 50; ++k) s = fmaf(h1[k], wr[k], s);
        orow[o] = tanhf(s) + h1[(o < 50) ? o : o - 50];
    }
}

// ---------------------------------------------------------------------------
// K4: Gram stage via f32 WMMA, fed from LDS.
// Stage the whole (128 x OSTRIDE) f32 tile (57 KB) into LDS with CDNA5 async
// copies (GLOBAL_LOAD_ASYNC_TO_LDS_B128, tracked by ASYNCcnt), then per (b,n):
// G = 2 * out^T (K=128, zero-padded row 127) x out[:, :16].
// Wave w handles M-tile rows [16w, 16w+16); 32 chained V_WMMA_F32_16X16X4_F32.
// A (16x4, M x K): lane<16 -> M=lane, vgprs = K{2h, 2h+1} with h = lane>>4.
// B (4x16, K x N): lane<16 -> N=lane, vgprs = K{2h, 2h+1}.
// D: vgpr r -> M = 16w + r + 8h, N = lane&15.
// ---------------------------------------------------------------------------
__global__ __launch_bounds__(256) void k_gram(
    const float* __restrict__ outbuf,  // (B*NC, 128, OSTRIDE)
    float* __restrict__ dout)          // (B*NC, OUTC)
{
    __shared__ float tile[128 * OSTRIDE];          // 57,344 bytes

    int bn   = blockIdx.x;
    int wid  = threadIdx.x >> 5;
    int lane = threadIdx.x & 31;

    // ---- async stage: 3584 x 16B chunks, 14 per thread --------------------
    const float* gbase = outbuf + (size_t)bn * 128 * OSTRIDE;
    unsigned int  lbase = (unsigned int)(uintptr_t)tile;   // LDS addr = flat[31:0]
    #pragma unroll
    for (int it = 0; it < 14; ++it) {
        int c = threadIdx.x + it * 256;                    // chunk index
        unsigned long long ga = (unsigned long long)(uintptr_t)(gbase + c * 4);
        unsigned int       la = lbase + (unsigned int)c * 16u;
        asm volatile("global_load_async_to_lds_b128 %0, %1, off"
                     :: "v"(la), "v"(ga) : "memory");
    }
    asm volatile("s_wait_asynccnt 0x0" ::: "memory");
    __syncthreads();

    if (wid >= 7) return;              // uniform per wave; 7 tiles cover M=0..111

    int half = lane >> 4;
    int l    = lane & 15;
    int m0   = wid * 16;

    v8f acc = {};
    for (int k4 = 0; k4 < 128; k4 += 4) {
        int kA = k4 + half * 2;
        const float* r0 = &tile[kA * OSTRIDE];
        const float* r1 = r0 + OSTRIDE;
        v2f a;  a.x = r0[m0 + l]; a.y = r1[m0 + l];
        v2f bb; bb.x = r0[l];     bb.y = r1[l];
        acc = __builtin_amdgcn_wmma_f32_16x16x4_f32(
            /*neg_a=*/false, a, /*neg_b=*/false, bb,
            /*c_mod=*/(short)0, acc, /*reuse_a=*/false, /*reuse_b=*/false);
    }

    float* g = dout + (size_t)bn * OUTC + 100;
    #pragma unroll
    for (int r = 0; r < 8; ++r) {
        int row = m0 + r + 8 * half;
        if (row < 100) g[row * 16 + l] = 2.0f * acc[r];
    }
}

// ---------------------------------------------------------------------------
extern "C" void kernel_launch(void* const* d_in, const int* in_sizes, int n_in,
                              void* d_out, int out_size, void* d_ws, size_t ws_size,
                              hipStream_t stream) {
    const float* ac   = (const float*)d_in[0];   // atom_coords
    const float* cc   = (const float*)d_in[1];   // charge_coords
    const float* q    = (const float*)d_in[2];   // charges
    const float* eW0  = (const float*)d_in[3];
    const float* eb0  = (const float*)d_in[4];
    const float* eW1  = (const float*)d_in[5];
    const float* eb1  = (const float*)d_in[6];
    const float* eW2  = (const float*)d_in[7];
    const float* eb2  = (const float*)d_in[8];
    const float* fW0  = (const float*)d_in[9];
    const float* fb0  = (const float*)d_in[10];
    const float* fW1  = (const float*)d_in[11];
    const float* fb1  = (const float*)d_in[12];
    const float* fW2  = (const float*)d_in[13];
    const float* fb2  = (const float*)d_in[14];
    const int*   types = (const int*)d_in[15];
    const unsigned char* mask = (const unsigned char*)d_in[16];

    float* out = (float*)d_out;
    float* ws  = (float*)d_ws;

    float* w_esp = ws;            // 512 floats
    float* w_ef  = ws + 512;      // 1536 floats
    float* w_out = ws + 2048;     // 512 * 128 * OSTRIDE floats (~29 MB, L2-resident)

    k_esp_field<<<B_ * NC_, 256, 0, stream>>>(ac, cc, q, mask, w_esp, w_ef);
    k_ef_mlp  <<<B_ * NC_, 128, 0, stream>>>(ac, w_ef, fW0, fb0, fW1, fb1, fW2, fb2, types, w_out);
    k_esp_mlp <<<B_ * NC_ / 128, 128, 0, stream>>>(w_esp, eW0, eb0, eW1, eb1, eW2, eb2, types, out);
    k_gram    <<<B_ * NC_, 256, 0, stream>>>(w_out, out);
}